// Feature_extra_specific_65257733095929
// MI455X (gfx1250) — compile-verified
//
#include <hip/hip_runtime.h>
#include <stdint.h>

typedef __attribute__((ext_vector_type(2))) float v2f;
typedef __attribute__((ext_vector_type(4))) float v4f;
typedef __attribute__((ext_vector_type(8))) float v8f;

#define BB 64
#define LL 512
#define DD 512
#define AA 8
#define K2D 1024            // 2*D (concat of H row and user row)
#define MTOT (BB * LL)      // 32768 rows of "Batch"
#define KC 128              // k-chunk staged in LDS
#define LDS_STRIDE (KC + 4) // pad -> conflict-free ds_load_b64 A-fragments

// ---------------------------------------------------------------------------
// Phase 1: for each aspect a:
//   Y = Batch @ W[a]  (both W_map and W_proj, sharing the Batch tile in LDS)
//   ws[a,m,n] = sigmoid(Ymap + bmap) * (Yproj + bproj)   (layout (A, B*L, D))
// Block = 256 threads = 8 waves. Block tile: 16 (m) x 128 (n), one aspect.
// X chunk is staged into LDS with CDNA5 async global->LDS DMA (ASYNCcnt).
// grid = (2048 m-tiles, 4 n-blocks, 8 aspects)
// ---------------------------------------------------------------------------
__global__ __launch_bounds__(256)
void gated_gemm_f32(const float* __restrict__ H,
                    const float* __restrict__ user,
                    const float* __restrict__ Wmap,
                    const float* __restrict__ bmap,
                    const float* __restrict__ Wproj,
                    const float* __restrict__ bproj,
                    float* __restrict__ ws)
{
    __shared__ float lA[16 * LDS_STRIDE];

    const int mt   = blockIdx.x;           // 0..2047
    const int nb   = blockIdx.y;           // 0..3
    const int a    = blockIdx.z;           // 0..7
    const int m0   = mt << 4;              // 16 rows per tile
    const int bidx = m0 >> 9;              // batch index (tile never straddles)
    const int tid  = threadIdx.x;
    const int lane = tid & 31;
    const int wave = tid >> 5;
    const int l15  = lane & 15;
    const int apair = (lane >> 4) << 1;    // k sub-offset 0 or 2 (A/B frag layout)
    const int n0   = (nb << 7) + (wave << 4);

    const float* __restrict__ Wm = Wmap  + (size_t)a * (K2D * DD);
    const float* __restrict__ Wp = Wproj + (size_t)a * (K2D * DD);

    // per-thread staging slot: 8 floats (= 2 x B128), same LDS/global row
    const int e = tid << 3;
    const int sr = e >> 7;                 // row in tile
    const int sc_ = e & 127;               // col in chunk
    // low 32 bits of the flat-cast shared pointer == LDS byte offset
    const unsigned ldsDst =
        (unsigned)(uintptr_t)(&lA[sr * LDS_STRIDE + sc_]);

    v8f cmap = {};
    v8f cproj = {};

    for (int k0 = 0; k0 < K2D; k0 += KC) {
        // ---- async DMA stage of X chunk: 16 rows x 128 k into LDS ----
        {
            const float* src = (k0 < DD)
                ? (H    + (size_t)(m0 + sr) * DD + (k0 + sc_))
                : (user + (size_t)bidx * DD + (k0 - DD + sc_));
            unsigned long long ga = (unsigned long long)(uintptr_t)src;
            // IOFFSET is added to BOTH the global src and LDS dst addresses,
            // so one address pair covers both 16B transfers.
            asm volatile(
                "global_load_async_to_lds_b128 %0, %1, off\n\t"
                "global_load_async_to_lds_b128 %0, %1, off offset:16"
                :: "v"(ldsDst), "v"(ga) : "memory");
            asm volatile("s_wait_asynccnt 0x0" ::: "memory");
        }
        __syncthreads();

        // prefetch next weight chunk (resident in L2; emits global_prefetch_b8)
        if (k0 + KC < K2D) {
            __builtin_prefetch(Wm + (size_t)(k0 + KC) * DD + n0 + l15, 0, 0);
            __builtin_prefetch(Wp + (size_t)(k0 + KC) * DD + n0 + l15, 0, 0);
        }

        const float* __restrict__ wmk = Wm + (size_t)k0 * DD + n0 + l15;
        const float* __restrict__ wpk = Wp + (size_t)k0 * DD + n0 + l15;

        #pragma unroll 4
        for (int kk = 0; kk < KC; kk += 4) {
            const int krow = kk + apair;
            // A fragment: lane holds X[m0 + l15][k0 + krow + {0,1}]
            v2f afrag = *(const v2f*)&lA[l15 * LDS_STRIDE + krow];
            // B fragments: lane holds W[a][k0 + krow + {0,1}][n0 + l15]
            v2f bm, bp;
            bm.x = wmk[(size_t)(krow)     * DD];
            bm.y = wmk[(size_t)(krow + 1) * DD];
            bp.x = wpk[(size_t)(krow)     * DD];
            bp.y = wpk[(size_t)(krow + 1) * DD];
            cmap  = __builtin_amdgcn_wmma_f32_16x16x4_f32(
                        false, afrag, false, bm, (short)0, cmap, false, false);
            cproj = __builtin_amdgcn_wmma_f32_16x16x4_f32(
                        false, afrag, false, bp, (short)0, cproj, false, false);
        }
        __syncthreads();
    }

    // ---- epilogue: bias, sigmoid gate, store to ws in (A, B*L, D) layout ----
    const int   nn    = n0 + l15;
    const float biasm = bmap [a * DD + nn];
    const float biasp = bproj[a * DD + nn];
    const int   mhi   = (lane >> 4) << 3;     // lanes 16..31 hold rows +8
    const size_t outbase = ((size_t)a * MTOT + m0) * DD + nn;

    #pragma unroll
    for (int v = 0; v < 8; ++v) {
        const float g = 1.0f / (1.0f + __expf(-(cmap[v] + biasm)));
        const float p = cproj[v] + biasp;
        ws[outbase + (size_t)(v + mhi) * DD] = g * p;
    }
}

// ---------------------------------------------------------------------------
// Phase 2: one block per (b, a).  item_asp[b,l,a,:] is the flat buffer at
// ((b*L + l)*A + a)*D (the reference's raw reshape).  scores -> softmax over
// L -> weighted sum over L -> out[b,a,:].
// ---------------------------------------------------------------------------
__global__ __launch_bounds__(256)
void attn_reduce(const float* __restrict__ ws,
                 const float* __restrict__ asp,
                 float* __restrict__ out)
{
    const int b    = blockIdx.x >> 3;
    const int a    = blockIdx.x & 7;
    const int tid  = threadIdx.x;
    const int lane = tid & 31;
    const int wave = tid >> 5;

    __shared__ float sv[DD];    // asp_vs_d[a,:]
    __shared__ float sc[LL];    // scores -> exp values
    __shared__ float red[8];

    sv[tid]       = asp[a * DD + tid];
    sv[tid + 256] = asp[a * DD + tid + 256];
    __syncthreads();

    // ---- scores: wave w handles l = w*64 .. w*64+63 ----
    const float* __restrict__ base = ws + ((size_t)(b * LL) * AA + a) * DD;
    for (int i = 0; i < 64; ++i) {
        const int l = (wave << 6) + i;
        const float* __restrict__ row = base + (size_t)l * (AA * DD);
        float s = 0.0f;
        #pragma unroll
        for (int j = 0; j < 16; ++j) {
            const int d = (j << 5) + lane;
            s += row[d] * sv[d];
        }
        #pragma unroll
        for (int off = 16; off; off >>= 1) s += __shfl_xor(s, off, 32);
        if (lane == 0) sc[l] = s + 1e-24f;
    }
    __syncthreads();

    // ---- softmax over L=512 (each thread owns entries tid, tid+256) ----
    float m1 = fmaxf(sc[tid], sc[tid + 256]);
    #pragma unroll
    for (int off = 16; off; off >>= 1) m1 = fmaxf(m1, __shfl_xor(m1, off, 32));
    if (lane == 0) red[wave] = m1;
    __syncthreads();
    float mx = fmaxf(fmaxf(fmaxf(red[0], red[1]), fmaxf(red[2], red[3])),
                     fmaxf(fmaxf(red[4], red[5]), fmaxf(red[6], red[7])));

    const float e0 = __expf(sc[tid]       - mx);
    const float e1 = __expf(sc[tid + 256] - mx);
    float ssum = e0 + e1;
    #pragma unroll
    for (int off = 16; off; off >>= 1) ssum += __shfl_xor(ssum, off, 32);
    __syncthreads();                 // protect red[] reuse
    if (lane == 0) red[wave] = ssum;
    sc[tid]       = e0;              // each thread rewrites only its own entries
    sc[tid + 256] = e1;
    __syncthreads();
    const float tot = red[0] + red[1] + red[2] + red[3] +
                      red[4] + red[5] + red[6] + red[7];
    const float inv = 1.0f / tot;

    // ---- weighted sum over l (coalesced across n) ----
    float acc0 = 0.0f, acc1 = 0.0f;
    for (int l = 0; l < LL; ++l) {
        const float* __restrict__ row = base + (size_t)l * (AA * DD);
        const float w = sc[l];
        acc0 += w * row[tid];
        acc1 += w * row[tid + 256];
    }
    const size_t ob = ((size_t)b * AA + a) * DD;
    out[ob + tid]       = acc0 * inv;
    out[ob + tid + 256] = acc1 * inv;
}

// ---------------------------------------------------------------------------
extern "C" void kernel_launch(void* const* d_in, const int* in_sizes, int n_in,
                              void* d_out, int out_size, void* d_ws, size_t ws_size,
                              hipStream_t stream) {
    const float* H     = (const float*)d_in[0];   // (B, L, D)
    const float* ue    = (const float*)d_in[1];   // (B, D)
    const float* asp   = (const float*)d_in[2];   // (A, D)
    const float* Wmap  = (const float*)d_in[3];   // (A, 2D, D)
    const float* bmap  = (const float*)d_in[4];   // (A, D)
    const float* Wproj = (const float*)d_in[5];   // (A, 2D, D)
    const float* bproj = (const float*)d_in[6];   // (A, D)
    float* out = (float*)d_out;                   // (B, A, D)
    float* ws  = (float*)d_ws;                    // map_asp (A, B*L, D) f32

    dim3 g1(MTOT / 16, DD / 128, AA);             // 2048 x 4 x 8
    gated_gemm_f32<<<g1, 256, 0, stream>>>(H, ue, Wmap, bmap, Wproj, bproj, ws);

    attn_reduce<<<dim3(BB * AA), 256, 0, stream>>>(ws, asp, out);
}